// EGATLayer_64596308132188
// MI455X (gfx1250) — compile-verified
//
#include <hip/hip_runtime.h>

typedef float v2f __attribute__((ext_vector_type(2)));
typedef float v8f __attribute__((ext_vector_type(8)));

#define N_NODES 100000
#define E_EDGES 1600000
#define CSTRIDE 196   // LDS row stride for 192-wide combined tile (196 mod 64 == 4 -> conflict-free)

// ---------------------------------------------------------------------------
// Kernel 1: node projections  x_source = x@Wsrc.T+bsrc, x_target = x@Wtgt.T+btgt
// One wave handles a 16-row tile; fp32 WMMA 16x16x4, K=64.
// Also initializes out = x_target.
// ---------------------------------------------------------------------------
__global__ __launch_bounds__(64) void egat_node_proj(
    const float* __restrict__ x,
    const float* __restrict__ Wsrc, const float* __restrict__ bsrc,
    const float* __restrict__ Wtgt, const float* __restrict__ btgt,
    float* __restrict__ xsrc_ws, float* __restrict__ xtgt_ws,
    float* __restrict__ out)
{
    __shared__ float s_x[2][16 * 68];          // 68-float stride -> conflict-free A frags
    const int lane = threadIdx.x & 31;
    const int w    = threadIdx.x >> 5;
    const int tile = blockIdx.x * 2 + w;       // 16-row tile index (6250 total, exact)
    const int rbase = tile * 16;

    float* xt = s_x[w];
    // stage 16x64 tile of x into LDS (float4 granules)
    for (int i = lane; i < 16 * 16; i += 32) {
        int m = i >> 4, c4 = (i & 15) << 2;
        const float4 v = *reinterpret_cast<const float4*>(x + (size_t)(rbase + m) * 64 + c4);
        *reinterpret_cast<float4*>(&xt[m * 68 + c4]) = v;
    }
    // LDS ops are in-order within a wave; no barrier needed for own region.

    const int mrow = lane & 15;
    const int hi   = lane >> 4;          // 0 / 1
    const int koff = hi << 1;            // K pair offset 0 / 2 (ISA A/B f32 layout)

    for (int proj = 0; proj < 2; ++proj) {
        const float* W    = proj ? Wtgt : Wsrc;
        const float* bias = proj ? btgt : bsrc;
        float* dst        = proj ? xtgt_ws : xsrc_ws;
        for (int nt = 0; nt < 4; ++nt) {
            const int n = nt * 16 + mrow;          // output column for this lane
            v8f acc = {};
            for (int kc = 0; kc < 64; kc += 4) {
                v2f a = *reinterpret_cast<const v2f*>(&xt[mrow * 68 + kc + koff]);
                v2f b = *reinterpret_cast<const v2f*>(W + (size_t)n * 64 + kc + koff);
                acc = __builtin_amdgcn_wmma_f32_16x16x4_f32(
                        false, a, false, b, (short)0, acc, false, false);
            }
            const float bn = bias[n];
#pragma unroll
            for (int r = 0; r < 8; ++r) {          // D vgpr r -> rows r (lo half) / r+8 (hi half)
                const int row = rbase + r + (hi << 3);
                const float v = acc[r] + bn;
                dst[(size_t)row * 64 + n] = v;
                if (proj) out[(size_t)row * 64 + n] = v;   // out seeded with x_target
            }
        }
    }
}

// ---------------------------------------------------------------------------
// Kernel 2: per-edge attention MLP (fp32 WMMA) + scatter-add of messages.
// One wave = one 16-edge tile. 4 waves / block, 25,000 blocks (exact).
// ---------------------------------------------------------------------------
__global__ __launch_bounds__(128) void egat_edge_mlp(
    const float* __restrict__ edge_attr, const int* __restrict__ eidx,
    const float* __restrict__ We,  const float* __restrict__ be,
    const float* __restrict__ W1,  const float* __restrict__ bias1,
    const float* __restrict__ W2,  const float* __restrict__ bias2,
    const float* __restrict__ W3,  const float* __restrict__ bias3,
    const float* __restrict__ xsrc, const float* __restrict__ xtgt,
    float* __restrict__ out)
{
    __shared__ float s_comb[4][16 * CSTRIDE];   // 50176 B: combined tile, later reused for h1/h2
    __shared__ float s_We[64 * 8];
    __shared__ float s_be[64];
    __shared__ float s_b1[64];
    __shared__ float s_b2[32];
    __shared__ float s_W3[32];
    __shared__ float s_ea[4][16 * 8];
    __shared__ int   s_src[4][16];
    __shared__ int   s_tgt[4][16];
    __shared__ float s_e[4][16];

    const int tid = threadIdx.x;
    // block-wide weight staging
    for (int i = tid; i < 512; i += 128) s_We[i] = We[i];
    if (tid < 64) { s_be[tid] = be[tid]; s_b1[tid] = bias1[tid]; }
    if (tid < 32) { s_b2[tid] = bias2[tid]; s_W3[tid] = W3[tid]; }
    __syncthreads();

    const int lane  = tid & 31;
    const int w     = tid >> 5;
    const int tile  = blockIdx.x * 4 + w;       // 16-edge tile (100,000 total, exact)
    const int ebase = tile * 16;

    // ---- Phase A: indices + edge_attr tile ----
    if (lane < 16) {
        s_src[w][lane] = eidx[ebase + lane];
        s_tgt[w][lane] = eidx[E_EDGES + ebase + lane];
    }
    for (int i = lane; i < 16 * 8; i += 32)
        s_ea[w][i] = edge_attr[(size_t)ebase * 8 + i];

    float* comb = s_comb[w];

    // ---- Phase B: gather x_source (cols 0..63), x_target (cols 64..127) ----
    for (int m = 0; m < 16; ++m) {
        const int s = s_src[w][m];
        const int t = s_tgt[w][m];
        for (int c = lane; c < 64; c += 32) {
            comb[m * CSTRIDE + c]      = xsrc[(size_t)s * 64 + c];
            comb[m * CSTRIDE + 64 + c] = xtgt[(size_t)t * 64 + c];
        }
    }
    // edge projection -> cols 128..191 (K=8, VALU; tiny vs the MLP GEMMs)
    for (int i = lane; i < 16 * 64; i += 32) {
        const int m = i >> 6, j = i & 63;
        float acc = s_be[j];
#pragma unroll
        for (int d = 0; d < 8; ++d)
            acc += s_ea[w][m * 8 + d] * s_We[j * 8 + d];
        comb[m * CSTRIDE + 128 + j] = acc;
    }

    const int mrow = lane & 15;
    const int hi   = lane >> 4;
    const int koff = hi << 1;

    // ---- Layer 1: [16x192] @ W1.T -> [16x64], single K sweep, 4 live accumulators ----
    v8f a0 = {}, a1 = {}, a2 = {}, a3 = {};
    for (int kc = 0; kc < 192; kc += 4) {
        v2f av = *reinterpret_cast<const v2f*>(&comb[mrow * CSTRIDE + kc + koff]);
        v2f w0 = *reinterpret_cast<const v2f*>(W1 + (size_t)(0 * 16 + mrow) * 192 + kc + koff);
        v2f w1 = *reinterpret_cast<const v2f*>(W1 + (size_t)(1 * 16 + mrow) * 192 + kc + koff);
        v2f w2 = *reinterpret_cast<const v2f*>(W1 + (size_t)(2 * 16 + mrow) * 192 + kc + koff);
        v2f w3 = *reinterpret_cast<const v2f*>(W1 + (size_t)(3 * 16 + mrow) * 192 + kc + koff);
        a0 = __builtin_amdgcn_wmma_f32_16x16x4_f32(false, av, false, w0, (short)0, a0, false, false);
        a1 = __builtin_amdgcn_wmma_f32_16x16x4_f32(false, av, false, w1, (short)0, a1, false, false);
        a2 = __builtin_amdgcn_wmma_f32_16x16x4_f32(false, av, false, w2, (short)0, a2, false, false);
        a3 = __builtin_amdgcn_wmma_f32_16x16x4_f32(false, av, false, w3, (short)0, a3, false, false);
    }
    // bias + leaky-relu, write h1 into comb cols 0..63 (combined no longer needed)
#pragma unroll
    for (int r = 0; r < 8; ++r) {
        const int m = r + (hi << 3);
        float h;
        h = a0[r] + s_b1[ 0 + mrow]; h = h > 0.f ? h : 0.01f * h; comb[m * CSTRIDE +  0 + mrow] = h;
        h = a1[r] + s_b1[16 + mrow]; h = h > 0.f ? h : 0.01f * h; comb[m * CSTRIDE + 16 + mrow] = h;
        h = a2[r] + s_b1[32 + mrow]; h = h > 0.f ? h : 0.01f * h; comb[m * CSTRIDE + 32 + mrow] = h;
        h = a3[r] + s_b1[48 + mrow]; h = h > 0.f ? h : 0.01f * h; comb[m * CSTRIDE + 48 + mrow] = h;
    }

    // ---- Layer 2: [16x64] @ W2.T -> [16x32] ----
    v8f c0 = {}, c1 = {};
    for (int kc = 0; kc < 64; kc += 4) {
        v2f av = *reinterpret_cast<const v2f*>(&comb[mrow * CSTRIDE + kc + koff]);
        v2f w0 = *reinterpret_cast<const v2f*>(W2 + (size_t)(mrow) * 64 + kc + koff);
        v2f w1 = *reinterpret_cast<const v2f*>(W2 + (size_t)(16 + mrow) * 64 + kc + koff);
        c0 = __builtin_amdgcn_wmma_f32_16x16x4_f32(false, av, false, w0, (short)0, c0, false, false);
        c1 = __builtin_amdgcn_wmma_f32_16x16x4_f32(false, av, false, w1, (short)0, c1, false, false);
    }
#pragma unroll
    for (int r = 0; r < 8; ++r) {
        const int m = r + (hi << 3);
        float h;
        h = c0[r] + s_b2[ 0 + mrow]; h = h > 0.f ? h : 0.01f * h; comb[m * CSTRIDE + 64 + mrow] = h;
        h = c1[r] + s_b2[16 + mrow]; h = h > 0.f ? h : 0.01f * h; comb[m * CSTRIDE + 80 + mrow] = h;
    }

    // ---- Layer 3: [16x32] @ W3.T -> [16], relu ----
    if (lane < 16) {
        float acc = bias3[0];
#pragma unroll
        for (int j = 0; j < 32; ++j)
            acc += comb[lane * CSTRIDE + 64 + j] * s_W3[j];
        s_e[w][lane] = acc > 0.f ? acc : 0.f;
    }

    // ---- Scatter: out[tgt] += e * x_source[src] (hardware f32 atomics) ----
    for (int m = 0; m < 16; ++m) {
        const float e = s_e[w][m];                 // wave-uniform
        if (e != 0.f) {
            const int s = s_src[w][m];
            const int t = s_tgt[w][m];
#pragma unroll
            for (int half = 0; half < 2; ++half) {
                const int c = lane + 32 * half;
                const float v = e * xsrc[(size_t)s * 64 + c];
                unsafeAtomicAdd(&out[(size_t)t * 64 + c], v);
            }
        }
    }
}

// ---------------------------------------------------------------------------
extern "C" void kernel_launch(void* const* d_in, const int* in_sizes, int n_in,
                              void* d_out, int out_size, void* d_ws, size_t ws_size,
                              hipStream_t stream) {
    const float* x    = (const float*)d_in[0];
    const int*   eidx = (const int*)  d_in[1];
    const float* ea   = (const float*)d_in[2];
    const float* Wsrc = (const float*)d_in[3];
    const float* bsrc = (const float*)d_in[4];
    const float* Wtgt = (const float*)d_in[5];
    const float* btgt = (const float*)d_in[6];
    const float* We   = (const float*)d_in[7];
    const float* be   = (const float*)d_in[8];
    const float* W1   = (const float*)d_in[9];
    const float* b1   = (const float*)d_in[10];
    const float* W2   = (const float*)d_in[11];
    const float* b2   = (const float*)d_in[12];
    const float* W3   = (const float*)d_in[13];
    const float* b3   = (const float*)d_in[14];
    float* out  = (float*)d_out;

    float* xsrc = (float*)d_ws;                          // [N,64] fp32
    float* xtgt = xsrc + (size_t)N_NODES * 64;           // [N,64] fp32 (51.2 MB total)

    // 6250 16-row tiles, 2 waves/block -> 3125 blocks (exact)
    egat_node_proj<<<N_NODES / 32, 64, 0, stream>>>(
        x, Wsrc, bsrc, Wtgt, btgt, xsrc, xtgt, out);

    // 100,000 16-edge tiles, 4 waves/block -> 25,000 blocks (exact)
    egat_edge_mlp<<<E_EDGES / 64, 128, 0, stream>>>(
        ea, eidx, We, be, W1, b1, W2, b2, W3, b3, xsrc, xtgt, out);
}